// XSimGCL_60868276519165
// MI455X (gfx1250) — compile-verified
//
#include <hip/hip_runtime.h>
#include <hip/hip_bf16.h>
#include <cstdint>

typedef float v2f __attribute__((ext_vector_type(2)));
typedef float v8f __attribute__((ext_vector_type(8)));

#define EMBED_DIM 64
#define NUM_LAYERS 3

// cur <- concat(user_emb, item_emb); next <- 0; acc <- 0   (float4-vectorized)
__global__ void k_init(const float4* __restrict__ ue, const float4* __restrict__ ie,
                       float4* __restrict__ cur, float4* __restrict__ nxt,
                       float4* __restrict__ acc, int nu4, int total4) {
  int i = blockIdx.x * blockDim.x + threadIdx.x;
  if (i >= total4) return;
  cur[i] = (i < nu4) ? ue[i] : ie[i - nu4];
  nxt[i] = make_float4(0.f, 0.f, 0.f, 0.f);
  acc[i] = make_float4(0.f, 0.f, 0.f, 0.f);
}

// COO SpMM: y[rows[e]] += vals[e] * x[cols[e]]
// 16 lanes per edge, each lane owns a float4 slice (global_load_b128 gather +
// 4x global_atomic_add_f32 scatter). Rows are 256B-aligned -> 1 cacheline/edge.
__global__ void k_spmm(const float* __restrict__ x, float* __restrict__ y,
                       const float* __restrict__ vals, const int* __restrict__ rows,
                       const int* __restrict__ cols, int nnz) {
  long long t = (long long)blockIdx.x * blockDim.x + threadIdx.x;
  int e = (int)(t >> 4);
  int lane = (int)(t & 15);
  if (e >= nnz) return;
  int r = rows[e];
  int c = cols[e];
  float v = vals[e];
  const float4 xv = *(const float4*)(x + (size_t)c * EMBED_DIM + lane * 4);
  float* yp = y + (size_t)r * EMBED_DIM + lane * 4;
  atomicAdd(yp + 0, v * xv.x);
  atomicAdd(yp + 1, v * xv.y);
  atomicAdd(yp + 2, v * xv.z);
  atomicAdd(yp + 3, v * xv.w);
}

// acc += nxt; zero the buffer that will receive the next layer's scatter
__global__ void k_accum_zero(float4* __restrict__ acc, const float4* __restrict__ nxt,
                             float4* __restrict__ zt, int total4) {
  int i = blockIdx.x * blockDim.x + threadIdx.x;
  if (i >= total4) return;
  float4 a = acc[i];
  float4 n = nxt[i];
  acc[i] = make_float4(a.x + n.x, a.y + n.y, a.z + n.z, a.w + n.w);
  zt[i]  = make_float4(0.f, 0.f, 0.f, 0.f);
}

// Batched dot via V_WMMA_F32_16X16X4_F32: one wave handles 16 (u,i) pairs.
// A = U tile (16x64), B = I^T tile (64x16); D = U * I^T, diagonal = dots.
// f32 A 16x4 layout : lane L holds row  M=L&15, VGPR j = K=(L>>4)*2+j
// f32 B 4x16 layout : lane L holds col  N=L&15, VGPR j = K=(L>>4)*2+j
// -> both operands are the same 8-byte load pattern from the gathered rows.
__global__ void k_dot_wmma(const float* __restrict__ acc, const int* __restrict__ uid,
                           const int* __restrict__ iid, int num_users,
                           float* __restrict__ out, int batch, float scale) {
  int gt = blockIdx.x * blockDim.x + threadIdx.x;
  int wave = gt >> 5;
  int lane = gt & 31;
  int base = wave * 16;
  if (base >= batch) return;           // wave-uniform: EXEC stays all-ones
  int m = lane & 15;
  int koff = (lane >> 4) << 1;         // 0 or 2
  int b = base + m;
  if (b >= batch) b = batch - 1;       // clamp (keeps EXEC full for WMMA)
  const float* urow = acc + (size_t)uid[b] * EMBED_DIM;
  const float* irow = acc + ((size_t)iid[b] + (size_t)num_users) * EMBED_DIM;

  v8f c = {};
#pragma unroll
  for (int k0 = 0; k0 < EMBED_DIM; k0 += 4) {
    v2f a  = *(const v2f*)(urow + k0 + koff);   // 8B aligned: k0%4==0, koff even
    v2f bm = *(const v2f*)(irow + k0 + koff);
    c = __builtin_amdgcn_wmma_f32_16x16x4_f32(
        /*neg_a=*/false, a, /*neg_b=*/false, bm,
        /*c_mod=*/(short)0, c, /*reuse_a=*/false, /*reuse_b=*/false);
  }

  // Extract diagonal D[m][m]:
  //   m in 0..7  -> lane m,     VGPR m
  //   m in 8..15 -> lane m+16,  VGPR m-8
  int sel = (lane < 16) ? lane : (lane - 24);
  float d = 0.f;
#pragma unroll
  for (int j = 0; j < 8; ++j)
    if (sel == j) d = c[j];

  bool valid = (lane < 8) || (lane >= 24);
  int outIdx = base + ((lane < 16) ? lane : (lane - 16));
  if (valid && outIdx < batch) out[outIdx] = d * scale;
}

extern "C" void kernel_launch(void* const* d_in, const int* in_sizes, int n_in,
                              void* d_out, int out_size, void* d_ws, size_t ws_size,
                              hipStream_t stream) {
  const float* user_emb = (const float*)d_in[0];
  const float* item_emb = (const float*)d_in[1];
  const float* vals     = (const float*)d_in[2];
  const int*   rows     = (const int*)d_in[3];
  const int*   cols     = (const int*)d_in[4];
  const int*   uid      = (const int*)d_in[5];
  const int*   iid      = (const int*)d_in[6];

  const int nu    = in_sizes[0] / EMBED_DIM;
  const int ni    = in_sizes[1] / EMBED_DIM;
  const int nnz   = in_sizes[2];
  const int batch = in_sizes[5];
  const int N     = nu + ni;

  const size_t embBytes = (size_t)N * EMBED_DIM * sizeof(float);
  char* base = (char*)d_ws;
  float* bufA = (float*)(base);
  float* bufB = (float*)(base + embBytes);
  float* acc  = (float*)(base + 2 * embBytes);
  float* cur = bufA;
  float* nxt = bufB;

  const int total4 = N * (EMBED_DIM / 4);
  const int nu4    = nu * (EMBED_DIM / 4);
  dim3 blk(256);

  k_init<<<(total4 + 255) / 256, blk, 0, stream>>>(
      (const float4*)user_emb, (const float4*)item_emb,
      (float4*)cur, (float4*)nxt, (float4*)acc, nu4, total4);

  for (int l = 0; l < NUM_LAYERS; ++l) {
    long long th = (long long)nnz * 16;
    int nb = (int)((th + 255) / 256);
    k_spmm<<<nb, blk, 0, stream>>>(cur, nxt, vals, rows, cols, nnz);
    k_accum_zero<<<(total4 + 255) / 256, blk, 0, stream>>>(
        (float4*)acc, (const float4*)nxt, (float4*)cur, total4);
    float* tmp = cur; cur = nxt; nxt = tmp;
  }

  // final = acc / NUM_LAYERS on both sides -> scale product by 1/NUM_LAYERS^2
  const float scale = 1.0f / (float)(NUM_LAYERS * NUM_LAYERS);
  int waves = (batch + 15) / 16;
  long long th = (long long)waves * 32;
  int nb = (int)((th + 255) / 256);
  k_dot_wmma<<<nb, blk, 0, stream>>>(acc, uid, iid, nu, (float*)d_out, batch, scale);
}